// NCT_82162724372482
// MI455X (gfx1250) — compile-verified
//
#include <hip/hip_runtime.h>

#define UNITS_PER_BLOCK 8
#define THREADS 256

// fast sigmoid: 1/(1+exp(-x)) -> v_exp_f32 + v_rcp_f32 (~2 ulp, fine vs f32 ref)
__device__ __forceinline__ float sigm(float x) {
  return __builtin_amdgcn_rcpf(1.0f + __expf(-x));
}

// low 32 bits of a generic pointer to LDS == DS byte offset on AMDGPU
__device__ __forceinline__ unsigned lds_lo32(const void* p) {
  return (unsigned)(unsigned long long)(uintptr_t)p;
}

// Copy one level (512 floats = 2KB) global -> LDS via CDNA5 async path.
// Lane l copies 16B at byte offset 16*l; offset: applies to BOTH lds and
// global address (CDNA5 ISA), so 4 instrs cover 2KB,
// each instruction perfectly coalesced (512B contiguous per wave).
__device__ __forceinline__ void async_level_copy(const float* gsrc, float* lbuf, int lane) {
  unsigned laddr = lds_lo32(lbuf) + (unsigned)(lane << 4);
  unsigned long long gaddr = (unsigned long long)(uintptr_t)gsrc + (unsigned)(lane << 4);
  asm volatile(
      "global_load_async_to_lds_b128 %0, %1, off th:TH_LOAD_NT\n\t"
      "global_load_async_to_lds_b128 %0, %1, off offset:512 th:TH_LOAD_NT\n\t"
      "global_load_async_to_lds_b128 %0, %1, off offset:1024 th:TH_LOAD_NT\n\t"
      "global_load_async_to_lds_b128 %0, %1, off offset:1536 th:TH_LOAD_NT"
      :: "v"(laddr), "v"(gaddr)
      : "memory");
}

__device__ __forceinline__ void wait_async_le4() {
  asm volatile("s_wait_asynccnt 4" ::: "memory");
}
__device__ __forceinline__ void wait_async_0() {
  asm volatile("s_wait_asynccnt 0" ::: "memory");
}

// Each lane reads its contiguous 64B chunk from LDS (4x ds_load_b128) and
// applies sigmoid.
__device__ __forceinline__ void load_sig16(const float* buf, int lane, float v[16]) {
  const float4* p = reinterpret_cast<const float4*>(buf + (lane << 4));
  float4 a = p[0], b = p[1], c = p[2], d = p[3];
  v[0] = sigm(a.x);  v[1] = sigm(a.y);  v[2] = sigm(a.z);  v[3] = sigm(a.w);
  v[4] = sigm(b.x);  v[5] = sigm(b.y);  v[6] = sigm(b.z);  v[7] = sigm(b.w);
  v[8] = sigm(c.x);  v[9] = sigm(c.y);  v[10] = sigm(c.z); v[11] = sigm(c.w);
  v[12] = sigm(d.x); v[13] = sigm(d.y); v[14] = sigm(d.z); v[15] = sigm(d.w);
}

// One wave32 per (batch,tree) unit. Levels processed 9 -> 0 with a 2-deep
// async double buffer; deep levels fold lane-locally, shallow levels use
// shfl_xor butterflies for cross-lane segment means.
__global__ __launch_bounds__(THREADS) void NCT_tree_kernel(const float* __restrict__ x,
                                                           float* __restrict__ out,
                                                           int n_units) {
  __shared__ __align__(16) float smem[UNITS_PER_BLOCK * 1024];  // 2 x 512 floats per wave
  const int lane = threadIdx.x & 31;
  const int wave = threadIdx.x >> 5;
  const int u = blockIdx.x * UNITS_PER_BLOCK + wave;
  if (u >= n_units) return;  // wave-uniform, no barriers used

  const float* g = x + (size_t)u * 5120;  // 10 levels * 512 floats, contiguous
  float* buf0 = &smem[wave * 1024];
  float* buf1 = buf0 + 512;

  float t[16], v[16];

  // prefetch level 9 (leaves) into buf0
  async_level_copy(g + 9 * 512, buf0, lane);

  // ---- level 9: leaf factor is d1_9 itself (segment size 1)
  async_level_copy(g + 8 * 512, buf1, lane);
  wait_async_le4();
  load_sig16(buf0, lane, v);
#pragma unroll
  for (int m = 0; m < 16; ++m) t[m] = v[m];

  // ---- level 8: segment size 2, lane-local fold 16 -> 8
  async_level_copy(g + 7 * 512, buf0, lane);
  wait_async_le4();
  load_sig16(buf1, lane, v);
#pragma unroll
  for (int k = 0; k < 8; ++k) {
    float mu = (v[2 * k] + v[2 * k + 1]) * 0.5f;
    t[k] = fmaf(mu, t[2 * k + 1] - t[2 * k], t[2 * k]);  // (1-mu)*t0 + mu*t1
  }

  // ---- level 7: segment size 4, fold 8 -> 4
  async_level_copy(g + 6 * 512, buf1, lane);
  wait_async_le4();
  load_sig16(buf0, lane, v);
#pragma unroll
  for (int q = 0; q < 4; ++q) {
    float mu = ((v[4 * q] + v[4 * q + 1]) + (v[4 * q + 2] + v[4 * q + 3])) * 0.25f;
    t[q] = fmaf(mu, t[2 * q + 1] - t[2 * q], t[2 * q]);
  }

  // ---- level 6: segment size 8, fold 4 -> 2
  async_level_copy(g + 5 * 512, buf0, lane);
  wait_async_le4();
  load_sig16(buf1, lane, v);
#pragma unroll
  for (int h = 0; h < 2; ++h) {
    float s = 0.f;
#pragma unroll
    for (int j = 0; j < 8; ++j) s += v[8 * h + j];
    float mu = s * 0.125f;
    t[h] = fmaf(mu, t[2 * h + 1] - t[2 * h], t[2 * h]);
  }

  // ---- level 5: segment size 16 (exactly one lane), fold 2 -> 1
  async_level_copy(g + 4 * 512, buf1, lane);
  wait_async_le4();
  load_sig16(buf0, lane, v);
  float S;
  {
    float s = 0.f;
#pragma unroll
    for (int j = 0; j < 16; ++j) s += v[j];
    float mu = s * (1.0f / 16.0f);
    S = fmaf(mu, t[1] - t[0], t[0]);
  }

  // ---- levels 4..0: segment sizes 32..512 -> cross-lane group means.
  // Decision bit at level d is bit (4-d) of the lane id; accumulate into P.
  float P = 1.0f;
#pragma unroll
  for (int d = 4; d >= 0; --d) {
    const float* cb = ((9 - d) & 1) ? buf1 : buf0;  // buffer holding level d
    float* nb = ((9 - d) & 1) ? buf0 : buf1;
    if (d > 0) {
      async_level_copy(g + (d - 1) * 512, nb, lane);
      wait_async_le4();
    } else {
      wait_async_0();
    }
    load_sig16(cb, lane, v);
    float s = 0.f;
#pragma unroll
    for (int j = 0; j < 16; ++j) s += v[j];
#pragma unroll
    for (int mask = 1; mask <= (16 >> d); mask <<= 1) s += __shfl_xor(s, mask, 32);
    float mu = s * (1.0f / (float)(512 >> d));
    int bit = (lane >> (4 - d)) & 1;
    P *= bit ? mu : (1.0f - mu);
  }

  // out[u] = sum over 32 lanes of P * S   (each lane covers 16 leaves)
  float c = P * S;
#pragma unroll
  for (int mask = 1; mask <= 16; mask <<= 1) c += __shfl_xor(c, mask, 32);
  if (lane == 0) out[u] = c;
}

extern "C" void kernel_launch(void* const* d_in, const int* in_sizes, int n_in,
                              void* d_out, int out_size, void* d_ws, size_t ws_size,
                              hipStream_t stream) {
  (void)in_sizes; (void)n_in; (void)d_ws; (void)ws_size;
  const float* x = (const float*)d_in[0];
  float* out = (float*)d_out;
  const int n_units = out_size;  // 1024 * 16
  const int blocks = (n_units + UNITS_PER_BLOCK - 1) / UNITS_PER_BLOCK;
  NCT_tree_kernel<<<blocks, THREADS, 0, stream>>>(x, out, n_units);
}